// Paste_29678224015944
// MI455X (gfx1250) — compile-verified
//
#include <hip/hip_runtime.h>
#include <cstdint>
#include <cstddef>

// Paste-with-blend: out = background, except inside the per-batch (h x w)
// window at (y[b], x[b]) where out = m*patch + (1-m)*background.
// Pure memory-bound op: ~855 MB moved -> ~37us roofline at 23.3 TB/s.

typedef float v4f __attribute__((ext_vector_type(4)));

static constexpr int Bn = 32, Cn = 3, Hn = 1024, Wn = 1024;
static constexpr int hn = 256, wn = 256;
static constexpr int ROWS = 4;           // rows handled per block

__global__ __launch_bounds__(256) void paste_blend_kernel(
    const float* __restrict__ bg,
    const float* __restrict__ patch,
    const int*   __restrict__ xs,
    const int*   __restrict__ ys,
    const float* __restrict__ mask,
    float*       __restrict__ out)
{
    // grid = (Hn/ROWS, Cn, Bn): b, c, row are block-uniform with no div/mod.
    __shared__ __align__(16) float sp[wn];   // staged patch row
    __shared__ __align__(16) float sm[wn];   // staged mask row

    const int c    = blockIdx.y;
    const int b    = blockIdx.z;
    const int row0 = blockIdx.x * ROWS;
    const int t    = threadIdx.x;            // 0..255 -> cols [4t, 4t+3]
    const int col  = t << 2;

    const int x = xs[b];                     // block-uniform -> s_load
    const int y = ys[b];

    const size_t imgBase  = ((size_t)(b * Cn + c)) * Hn * (size_t)Wn;
    const size_t pimgBase = ((size_t)(b * Cn + c)) * hn * (size_t)wn;

    // Issue all ROWS background row loads up front (128-bit, non-temporal)
    // so they are all in flight before any barrier in the blend path.
    v4f v[ROWS];
    #pragma unroll
    for (int r = 0; r < ROWS; ++r) {
        const size_t rowOff = imgBase + (size_t)(row0 + r) * Wn;
        v[r] = __builtin_nontemporal_load((const v4f*)(bg + rowOff) + t);
    }

    #pragma unroll
    for (int r = 0; r < ROWS; ++r) {
        const int row = row0 + r;
        const int py  = row - y;             // row inside patch, if in [0, hn)
        if (py >= 0 && py < hn) {            // block-uniform branch
            const size_t prow = pimgBase + (size_t)py * wn;
            __syncthreads();                 // LDS safe to overwrite (prev iter)
#if defined(__gfx1250__)
            // gfx1250 async Global->LDS DMA (ASYNCcnt): 64 lanes x 16B = full
            // 256-float row; always in-bounds, clipping happens at blend time.
            if (t < 64) {
                const float* g = patch + prow + (size_t)(t << 2);
                unsigned lds = (unsigned)(uintptr_t)&sp[t << 2];
                asm volatile("global_load_async_to_lds_b128 %0, %1, off"
                             :: "v"(lds), "v"(g) : "memory");
            } else if (t < 128) {
                const int u = t - 64;
                const float* g = mask + prow + (size_t)(u << 2);
                unsigned lds = (unsigned)(uintptr_t)&sm[u << 2];
                asm volatile("global_load_async_to_lds_b128 %0, %1, off"
                             :: "v"(lds), "v"(g) : "memory");
            }
            asm volatile("s_wait_asynccnt 0x0" ::: "memory");
            __syncthreads();                 // publish DMA'd rows to all waves
#else
            if (t < 64) {
                *(v4f*)&sp[t << 2] = *(const v4f*)(patch + prow + (size_t)(t << 2));
            } else if (t < 128) {
                const int u = t - 64;
                *(v4f*)&sm[u << 2] = *(const v4f*)(mask + prow + (size_t)(u << 2));
            }
            __syncthreads();
#endif
            #pragma unroll
            for (int e = 0; e < 4; ++e) {
                const int px = col + e - x;  // column inside patch
                if (px >= 0 && px < wn) {    // right-edge crop
                    const float m = sm[px];
                    const float p = sp[px];
                    v[r][e] = m * p + (1.0f - m) * v[r][e];
                }
            }
        }
        const size_t rowOff = imgBase + (size_t)row * Wn;
        __builtin_nontemporal_store(v[r], (v4f*)(out + rowOff) + t);
    }
}

extern "C" void kernel_launch(void* const* d_in, const int* in_sizes, int n_in,
                              void* d_out, int out_size, void* d_ws, size_t ws_size,
                              hipStream_t stream) {
    // setup_inputs() order: background, patch, x, y, mask
    const float* bg    = (const float*)d_in[0];
    const float* patch = (const float*)d_in[1];
    const int*   xs    = (const int*)d_in[2];
    const int*   ys    = (const int*)d_in[3];
    const float* mask  = (const float*)d_in[4];
    float* out = (float*)d_out;
    (void)in_sizes; (void)n_in; (void)out_size; (void)d_ws; (void)ws_size;

    dim3 grid(Hn / ROWS, Cn, Bn);            // (row-group, channel, batch)
    paste_blend_kernel<<<grid, dim3(256), 0, stream>>>(
        bg, patch, xs, ys, mask, out);
}